// PEGIModel_43095701848412
// MI455X (gfx1250) — compile-verified
//
#include <hip/hip_runtime.h>

// ---------------------------------------------------------------------------
// PEGI splat render for MI455X (gfx1250, wave32).
//
// img_c = (Sum_n  w_n*t_n * (op_n*color_nc)) / Sum_n t_n,
// t_n = exp(-beta*depth_n)*(w_n + 1e-8).  Single pass over gaussians.
// Composite = V_WMMA_F32_16X16X4_F32 (A = per-pixel coefficients,
// B = pre-expanded per-lane op*color image), coefficient math on VALU:
// 5 FMAs (quadratic form, pre-scaled by -0.5*log2(e)) + one v_exp_f32.
// ---------------------------------------------------------------------------

typedef __attribute__((ext_vector_type(2))) float v2f;
typedef __attribute__((ext_vector_type(8))) float v8f;

#define NG   512
#define HH   384
#define WW   384
#define NPIX (HH * WW)
#define BETA 8.0f

// ws layout:
//   wsP[i*8]          = {qA,qB,qC,qD,qE,qF,ed,ed*1e-8}   (NG*8 floats, 16 KB)
//   wsB[kb*64+lane*2] = B-vgpr image per WMMA step        (NG/4*64 floats, 32 KB)
//     B vgpr j at lane (half*16+c) = (c<3 ? op*color[4kb+2*half+j][c] : 0)

__global__ __launch_bounds__(256) void pegi_setup(
    const float* __restrict__ mu, const float* __restrict__ log_sigma,
    const float* __restrict__ theta, const float* __restrict__ depth,
    const float* __restrict__ normal, const float* __restrict__ albedo,
    const float* __restrict__ roughness, const float* __restrict__ opacity,
    const float* __restrict__ light, const float* __restrict__ ambient,
    float* __restrict__ wsP, float* __restrict__ wsB)
{
    int i = blockIdx.x * blockDim.x + threadIdx.x;
    if (i >= NG) return;

    float sx = __expf(log_sigma[2 * i + 0]) + 1e-6f;
    float sy = __expf(log_sigma[2 * i + 1]) + 1e-6f;
    float th = theta[i];
    float ct = __cosf(th), st = __sinf(th);
    float ed = __expf(-BETA * depth[i]);

    // rotation/sigma rows:  u = ra0*dx + ra1*dy,  v = rb0*dx + rb1*dy
    float ra0 = ct / sx, ra1 = st / sx;
    float rb0 = -st / sy, rb1 = ct / sy;

    // quadratic form u^2+v^2 expanded in (px,py), pre-scaled by -0.5*log2(e)
    const float s = -0.721347520444482f;          // -0.5 * log2(e)
    float a = ra0 * ra0 + rb0 * rb0;
    float b = 2.0f * (ra0 * ra1 + rb0 * rb1);
    float c = ra1 * ra1 + rb1 * rb1;
    float mx = mu[2 * i + 0], my = mu[2 * i + 1];
    float qA = s * a;
    float qB = s * b;
    float qC = s * c;
    float qD = s * (-(2.0f * a * mx + b * my));
    float qE = s * (-(2.0f * c * my + b * mx));
    float qF = s * (a * mx * mx + b * mx * my + c * my * my);

    // shading
    float lx = light[0], ly = light[1], lz = light[2];
    float il = rsqrtf(lx * lx + ly * ly + lz * lz);
    lx *= il; ly *= il; lz *= il;
    float nx = normal[3 * i], ny = normal[3 * i + 1], nz = normal[3 * i + 2];
    float in = rsqrtf(nx * nx + ny * ny + nz * nz);
    nx *= in; ny *= in; nz *= in;
    float ndotl = fmaxf(nx * lx + ny * ly + nz * lz, 0.0f);
    float hx = lx, hy = ly, hz = lz + 1.0f;               // + view (0,0,1)
    float ih = rsqrtf(hx * hx + hy * hy + hz * hz);
    hx *= ih; hy *= ih; hz *= ih;
    float sp = fmaxf(nx * hx + ny * hy + nz * hz, 0.0f);
    float s2 = sp * sp; s2 *= s2; s2 *= s2; s2 *= s2; s2 *= s2; // sp^32
    float specular = s2 * (1.0f - roughness[i]);
    float op = 1.0f / (1.0f + __expf(-opacity[i]));

    float oc[3];
    oc[0] = op * (albedo[3 * i + 0] * ndotl + specular + ambient[0]);
    oc[1] = op * (albedo[3 * i + 1] * ndotl + specular + ambient[1]);
    oc[2] = op * (albedo[3 * i + 2] * ndotl + specular + ambient[2]);

    float* P = wsP + i * 8;
    P[0] = qA; P[1] = qB; P[2] = qC; P[3] = qD;
    P[4] = qE; P[5] = qF; P[6] = ed; P[7] = ed * 1e-8f;

    // expand this gaussian's column of the B register image
    int kb   = i >> 2;            // WMMA step
    int r    = i & 3;             // K row within step
    int half = r >> 1;            // lane group (0: lanes 0-15, 1: 16-31)
    int j    = r & 1;             // B vgpr index (rows 0,2 -> b0; 1,3 -> b1)
    float* Bimg = wsB + kb * 64;
    #pragma unroll
    for (int cc = 0; cc < 16; ++cc) {
        int lane = half * 16 + cc;
        Bimg[lane * 2 + j] = (cc < 3) ? oc[cc] : 0.0f;
    }
}

// ---------------------------- main kernel ----------------------------------
// Block = 256 threads = 8 waves; each wave renders a 16-pixel tile.
// A (16x4 f32): lane L<16 -> (pixel L, K=0,1); lane L>=16 -> (pixel L-16, K=2,3).
__global__ __launch_bounds__(256) void pegi_main(
    const float* __restrict__ grid, const float* __restrict__ wsP,
    const float* __restrict__ wsB, float* __restrict__ out)
{
    __shared__ __align__(16) float sP[NG * 8];        // 16 KB
    __shared__ __align__(16) float sB[(NG / 4) * 64]; // 32 KB

    {   // cooperative LDS fill (b128 copies)
        const float4* gp4 = (const float4*)wsP;
        const float4* gb4 = (const float4*)wsB;
        float4* sp4 = (float4*)sP;
        float4* sb4 = (float4*)sB;
        for (int i = threadIdx.x; i < NG * 2; i += 256)     sp4[i] = gp4[i];
        for (int i = threadIdx.x; i < NG * 4; i += 256)     sb4[i] = gb4[i];
    }
    __syncthreads();

    const int lane = threadIdx.x & 31;
    const int wave = threadIdx.x >> 5;
    const int tile = blockIdx.x * 8 + wave;       // 16 pixels per tile
    const int col  = lane & 15;
    const int half = lane >> 4;

    const int pix = tile * 16 + col;
    const float px = grid[2 * pix + 0];
    const float py = grid[2 * pix + 1];

    v8f   acc  = {};       // D: 16 pixels x 16 cols (cols 0..2 = rgb)
    float sacc = 0.0f;     // per-lane partial of S

    const float* Bbase = sB + lane * 2;

    #pragma unroll 4
    for (int kb = 0; kb < NG / 4; ++kb) {
        const float* P0 = sP + (kb * 4 + half * 2) * 8;  // half-wave uniform
        const float* P1 = P0 + 8;                        //  -> LDS broadcast

        // q = qA px^2 + qB px py + qC py^2 + qD px + qE py + qF  (pre-scaled)
        float q0 = fmaf(px, fmaf(px, P0[0], fmaf(py, P0[1], P0[3])),
                            fmaf(py, fmaf(py, P0[2], P0[4]), P0[5]));
        float w0 = __builtin_amdgcn_exp2f(q0);           // v_exp_f32
        float t0 = fmaf(P0[6], w0, P0[7]);               // ed*(w+1e-8)

        float q1 = fmaf(px, fmaf(px, P1[0], fmaf(py, P1[1], P1[3])),
                            fmaf(py, fmaf(py, P1[2], P1[4]), P1[5]));
        float w1 = __builtin_amdgcn_exp2f(q1);
        float t1 = fmaf(P1[6], w1, P1[7]);

        sacc += t0 + t1;

        v2f A; A.x = w0 * t0; A.y = w1 * t1;

        v2f B = *(const v2f*)(Bbase + kb * 64);          // one ds_load_b64

        acc = __builtin_amdgcn_wmma_f32_16x16x4_f32(
            false, A, false, B, (short)0, acc, false, false);
    }

    // S per pixel: combine the two half-wave partials
    float stot = sacc + __shfl_xor(sacc, 16, 32);

    // D layout: vgpr j, lanes 0-15 -> pixel j; lanes 16-31 -> pixel j+8;
    // column (lane&15) -> channel.
    #pragma unroll
    for (int j = 0; j < 8; ++j) {
        float Sa = __shfl(stot, j, 32);
        float Sb = __shfl(stot, j + 8, 32);
        float S  = half ? Sb : Sa;
        float v  = acc[j] / S;
        v = fminf(fmaxf(v, 0.0f), 1.0f);
        int prow = tile * 16 + j + half * 8;
        if (col < 3) out[prow * 3 + col] = v;
    }
}

// ------------------------------ launcher -----------------------------------
extern "C" void kernel_launch(void* const* d_in, const int* in_sizes, int n_in,
                              void* d_out, int out_size, void* d_ws, size_t ws_size,
                              hipStream_t stream)
{
    const float* grid      = (const float*)d_in[0];
    const float* mu        = (const float*)d_in[1];
    const float* log_sigma = (const float*)d_in[2];
    const float* theta     = (const float*)d_in[3];
    const float* depth     = (const float*)d_in[4];
    const float* normal    = (const float*)d_in[5];
    const float* albedo    = (const float*)d_in[6];
    const float* roughness = (const float*)d_in[7];
    const float* opacity   = (const float*)d_in[8];
    const float* light     = (const float*)d_in[9];
    const float* ambient   = (const float*)d_in[10];

    float* wsP = (float*)d_ws;          // NG*8 floats      = 16 KB
    float* wsB = wsP + NG * 8;          // (NG/4)*64 floats = 32 KB

    pegi_setup<<<(NG + 255) / 256, 256, 0, stream>>>(
        mu, log_sigma, theta, depth, normal, albedo, roughness, opacity,
        light, ambient, wsP, wsB);

    pegi_main<<<NPIX / 128, 256, 0, stream>>>(
        grid, wsP, wsB, (float*)d_out);
}